// GCN_74491912782367
// MI455X (gfx1250) — compile-verified
//
#include <hip/hip_runtime.h>
#include <hip/hip_bf16.h>

typedef float v2f __attribute__((ext_vector_type(2)));
typedef float v8f __attribute__((ext_vector_type(8)));

#define NNODES 50000
#define NEDGES 600000
#define INDIM 128
#define HID 256
#define MTILES 5          // M-tiles per wave (80 rows per block; 50000 = 625*80)

// ---------------------------------------------------------------------------
// Degree / normalization
// ---------------------------------------------------------------------------
__global__ void zero_f32_kernel(float* __restrict__ p, int n) {
  int i = blockIdx.x * blockDim.x + threadIdx.x;
  if (i < n) p[i] = 0.0f;
}

__global__ void deg_accum_kernel(const int* __restrict__ dst, float* __restrict__ deg, int E) {
  int e = blockIdx.x * blockDim.x + threadIdx.x;
  if (e < E) atomicAdd(&deg[dst[e]], 1.0f);
}

__global__ void make_dinv_kernel(float* __restrict__ deg, int n) {
  int i = blockIdx.x * blockDim.x + threadIdx.x;
  if (i < n) deg[i] = rsqrtf(deg[i] + 1.0f);  // in-place: deg -> dinv
}

// ---------------------------------------------------------------------------
// Pack B[K,N] row-major into WMMA B-fragment order:
//   Bp[(k/2)*N*2 + n*2 + (k&1)] = B[k*N + n]
// so a lane's B fragment {B[k][n], B[k+1][n]} is one aligned b64 load.
// ---------------------------------------------------------------------------
__global__ void pack_b_kernel(const float* __restrict__ B, float* __restrict__ Bp,
                              int N, int total) {
  int i = blockIdx.x * blockDim.x + threadIdx.x;
  if (i >= total) return;
  int k = i / N;
  int n = i - k * N;
  Bp[((size_t)(k >> 1) * N + n) * 2 + (k & 1)] = B[i];
}

// ---------------------------------------------------------------------------
// fp32 WMMA GEMM + fused self-loop epilogue:
//   H[M,N]   = A[M,K] @ B[K,N]          (B pre-packed, see pack_b_kernel)
//   AGG[M,N] = H * dinv[row]^2          (initializes the aggregation buffer)
//
// Block = 128 threads (4 waves), __launch_bounds__(128,1) to unlock the full
// VGPR budget (5x4 tile = 160 accumulator VGPRs must stay in registers).
// Block b -> rows [80b, 80b+80). Wave w -> cols [64w, 64w+64).
// Each wave: 5 M-tiles x 4 N-tiles; each B fragment reused by 5 WMMAs.
//
// V_WMMA_F32_16X16X4_F32 operand layout (ISA 7.12.2):
//   A (16x4): lane L<16 holds A[L][k+0..1]; lane L+16 holds A[L][k+2..3]
//   B (4x16): lane L<16 holds B[k+0..1][L]; lane L+16 holds B[k+2..3][L]
//   C/D:      vgpr g, lanes 0-15 -> row g, col lane; lanes 16-31 -> row g+8
// ---------------------------------------------------------------------------
template <int K>
__global__ __launch_bounds__(128, 1)
void gemm_wmma_f32_kernel(const float* __restrict__ A,
                          const float* __restrict__ Bp,
                          const float* __restrict__ dinv,
                          float* __restrict__ H,
                          float* __restrict__ AGG,
                          int N) {
  const int wave = threadIdx.x >> 5;
  const int lane = threadIdx.x & 31;
  const int half = lane >> 4;    // selects K pair / row half
  const int lidx = lane & 15;    // M row (A) / N col (B)

  const int m0 = blockIdx.x * (16 * MTILES);
  const int n0 = wave * 64;

  v8f acc[MTILES][4] = {};

  // A base: row (m0+lidx), K-pair selected by half; tile t adds 16*K floats
  const float* abase = A + (size_t)(m0 + lidx) * K + 2 * half;
  // B base: packed pairs; k-step advances by N*2 floats per 2 k's
  const float* bbase = Bp + ((size_t)half * N + n0 + lidx) * 2;

  for (int k = 0; k < K; k += 4) {
    const float* bk = bbase + (size_t)(k >> 1) * N * 2;
    v2f bf[4];
    #pragma unroll
    for (int u = 0; u < 4; ++u)
      bf[u] = *(const v2f*)(bk + u * 32);        // 16 cols * 2 floats

    #pragma unroll
    for (int t = 0; t < MTILES; ++t) {
      v2f af = *(const v2f*)(abase + t * 16 * K + k);
      #pragma unroll
      for (int u = 0; u < 4; ++u) {
        acc[t][u] = __builtin_amdgcn_wmma_f32_16x16x4_f32(
            false, af, false, bf[u], (short)0, acc[t][u], false, false);
      }
    }
  }

  // Epilogue: store H and AGG = H * dinv^2 (self-loop term)
  #pragma unroll
  for (int t = 0; t < MTILES; ++t) {
    #pragma unroll
    for (int g = 0; g < 8; ++g) {
      const int row = m0 + t * 16 + g + 8 * half;
      const float di = dinv[row];
      const float sl = di * di;
      const size_t base = (size_t)row * N + n0 + lidx;
      #pragma unroll
      for (int u = 0; u < 4; ++u) {
        float v = acc[t][u][g];
        H[base + u * 16]   = v;
        AGG[base + u * 16] = v * sl;
      }
    }
  }
}

// ---------------------------------------------------------------------------
// Edge scatter: one wave per edge; each lane owns 8 contiguous floats
// (2 x b128 loads, coalesced 1KB/wave), then 8 f32 atomics.
// agg[dst] += h[src] * dinv[src]*dinv[dst]
// ---------------------------------------------------------------------------
__global__ void edge_scatter_kernel(const float* __restrict__ h,
                                    const int* __restrict__ src,
                                    const int* __restrict__ dst,
                                    const float* __restrict__ dinv,
                                    float* __restrict__ agg, int E) {
  int e = blockIdx.x * (blockDim.x >> 5) + (threadIdx.x >> 5);
  if (e >= E) return;
  int lane = threadIdx.x & 31;
  int s = src[e];
  int d = dst[e];
  float norm = dinv[s] * dinv[d];
  const float4* hs = (const float4*)(h + (size_t)s * HID) + lane * 2;
  float* ad = agg + (size_t)d * HID + lane * 8;
  float4 v0 = hs[0];
  float4 v1 = hs[1];
  atomicAdd(ad + 0, v0.x * norm);
  atomicAdd(ad + 1, v0.y * norm);
  atomicAdd(ad + 2, v0.z * norm);
  atomicAdd(ad + 3, v0.w * norm);
  atomicAdd(ad + 4, v1.x * norm);
  atomicAdd(ad + 5, v1.y * norm);
  atomicAdd(ad + 6, v1.z * norm);
  atomicAdd(ad + 7, v1.w * norm);
}

// ---------------------------------------------------------------------------
// out[i][f] = relu(agg[i][f] + b[f])
// ---------------------------------------------------------------------------
__global__ void bias_relu_kernel(const float* __restrict__ agg,
                                 const float* __restrict__ b,
                                 float* __restrict__ out, int total) {
  int i = blockIdx.x * blockDim.x + threadIdx.x;
  if (i < total) {
    float v = agg[i] + b[i & (HID - 1)];
    out[i] = v > 0.0f ? v : 0.0f;
  }
}

// ---------------------------------------------------------------------------
// out[node][o] = sum_k x[node][k] * Wl[k][o] + bl[o]   (OUT == 2)
// ---------------------------------------------------------------------------
__global__ void final_linear_kernel(const float* __restrict__ x,
                                    const float* __restrict__ Wl,
                                    const float* __restrict__ bl,
                                    float* __restrict__ out, int total) {
  int i = blockIdx.x * blockDim.x + threadIdx.x;
  if (i >= total) return;
  int node = i >> 1;
  int o = i & 1;
  const float* xr = x + (size_t)node * HID;
  float acc = bl[o];
  #pragma unroll 8
  for (int k = 0; k < HID; ++k) acc += xr[k] * Wl[k * 2 + o];
  out[i] = acc;
}

// ---------------------------------------------------------------------------
// Host orchestration
// ---------------------------------------------------------------------------
extern "C" void kernel_launch(void* const* d_in, const int* in_sizes, int n_in,
                              void* d_out, int out_size, void* d_ws, size_t ws_size,
                              hipStream_t stream) {
  const float* x   = (const float*)d_in[0];             // [N, 128]
  const int*   ei  = (const int*)d_in[1];               // [2, E]
  const float* W1  = (const float*)d_in[2];             // [128, 256]
  const float* b1  = (const float*)d_in[3];             // [256]
  const float* W2  = (const float*)d_in[4];             // [256, 256]
  const float* b2  = (const float*)d_in[5];             // [256]
  const float* Wl  = (const float*)d_in[6];             // [256, 2]
  const float* bl  = (const float*)d_in[7];             // [2]
  float*       out = (float*)d_out;                     // [N, 2]

  const int*   src = ei;
  const int*   dst = ei + NEDGES;

  // workspace layout (bytes)
  char* ws = (char*)d_ws;
  const size_t H_OFF   = 200192;                        // 50000*4 padded to 256
  const size_t HBYTES  = (size_t)NNODES * HID * 4;      // 51.2 MB
  const size_t AGG_OFF = H_OFF + HBYTES;
  const size_t X2_OFF  = AGG_OFF + HBYTES;
  const size_t BP_OFF  = X2_OFF + HBYTES;               // packed weights, <=256KB

  float* dinv = (float*)ws;
  float* H    = (float*)(ws + H_OFF);
  float* AGG  = (float*)(ws + AGG_OFF);
  float* X2   = (float*)(ws + X2_OFF);
  float* Bp   = (float*)(ws + BP_OFF);

  const int TPB = 256;
  const int nodeBlocks  = (NNODES + TPB - 1) / TPB;
  const int edgeBlocks  = (NEDGES + TPB - 1) / TPB;
  const int featTotal   = NNODES * HID;
  const int featBlocks  = (featTotal + TPB - 1) / TPB;
  const int waveEdgeBlk = (NEDGES + 7) / 8;             // 8 waves/block, 1 edge/wave
  const int gemmBlocks  = NNODES / (16 * MTILES);       // 625

  // ---- degree / dinv ----
  zero_f32_kernel<<<nodeBlocks, TPB, 0, stream>>>(dinv, NNODES);
  deg_accum_kernel<<<edgeBlocks, TPB, 0, stream>>>(dst, dinv, NEDGES);
  make_dinv_kernel<<<nodeBlocks, TPB, 0, stream>>>(dinv, NNODES);

  // ---- layer 1: H = x@W1 (+fused AGG init) ; scatter ; relu(agg+b1) -> X2 ----
  {
    const int tot = INDIM * HID;
    pack_b_kernel<<<(tot + TPB - 1) / TPB, TPB, 0, stream>>>(W1, Bp, HID, tot);
    gemm_wmma_f32_kernel<INDIM><<<gemmBlocks, 128, 0, stream>>>(x, Bp, dinv, H, AGG, HID);
    edge_scatter_kernel<<<waveEdgeBlk, TPB, 0, stream>>>(H, src, dst, dinv, AGG, NEDGES);
    bias_relu_kernel<<<featBlocks, TPB, 0, stream>>>(AGG, b1, X2, featTotal);
  }

  // ---- layer 2: H = X2@W2 (+fused AGG init) ; scatter ; relu(agg+b2) -> X2 ----
  {
    const int tot = HID * HID;
    pack_b_kernel<<<(tot + TPB - 1) / TPB, TPB, 0, stream>>>(W2, Bp, HID, tot);
    gemm_wmma_f32_kernel<HID><<<gemmBlocks, 128, 0, stream>>>(X2, Bp, dinv, H, AGG, HID);
    edge_scatter_kernel<<<waveEdgeBlk, TPB, 0, stream>>>(H, src, dst, dinv, AGG, NEDGES);
    bias_relu_kernel<<<featBlocks, TPB, 0, stream>>>(AGG, b2, X2, featTotal);
  }

  // ---- output projection ----
  const int outTotal = NNODES * 2;
  final_linear_kernel<<<(outTotal + TPB - 1) / TPB, TPB, 0, stream>>>(X2, Wl, bl, out, outTotal);
}